// DiscriminatorLoss_55611236548992
// MI455X (gfx1250) — compile-verified
//
#include <hip/hip_runtime.h>

// Ext-vector types: required for __builtin_nontemporal_load (needs vector/scalar,
// not the HIP_vector_type structs), and they lower to global_load_b128.
typedef __attribute__((ext_vector_type(4))) float f32x4;
typedef __attribute__((ext_vector_type(4))) int   i32x4;
typedef __attribute__((ext_vector_type(2))) float v2f;
typedef __attribute__((ext_vector_type(8))) float v8f;

#define NBLOCKS 2048
#define NTHREADS 256

// Exact fp32 wave32 sum using V_WMMA_F32_16X16X4_F32 with an all-ones B matrix.
// A layout (32-bit A, 16x4): lane L<16 -> VGPR0=A[L][0], VGPR1=A[L][1];
// lane L>=16 -> VGPR0=A[L-16][2], VGPR1=A[L-16][3].
// With a = {val, 0} and B = ones, D[m][n] = val(lane m) + val(lane m+16).
// Each lane's 8 D VGPRs then hold 8 of the 16 row-sums for its column;
// summing them and adding the partner half-wave's partial (shfl_xor 16)
// yields the full 32-lane sum in every lane. All products are x*1.0 -> exact.
__device__ __forceinline__ float wave_reduce_sum(float v) {
    v2f a; a.x = v;    a.y = 0.0f;
    v2f b; b.x = 1.0f; b.y = 1.0f;
    v8f c = {};
    c = __builtin_amdgcn_wmma_f32_16x16x4_f32(
            /*neg_a=*/false, a, /*neg_b=*/false, b,
            /*c_mod=*/(short)0, c, /*reuse_a=*/false, /*reuse_b=*/false);
    float p = ((c[0] + c[1]) + (c[2] + c[3])) + ((c[4] + c[5]) + (c[6] + c[7]));
    p += __shfl_xor(p, 16, 32);
    return p;
}

__global__ void __launch_bounds__(NTHREADS)
disc_loss_partial_kernel(const int* __restrict__ s,
                         const int* __restrict__ o,
                         const float* __restrict__ x,
                         float* __restrict__ partials,
                         int n4, int n) {
    const i32x4* s4 = (const i32x4*)s;
    const i32x4* o4 = (const i32x4*)o;
    const f32x4* x4 = (const f32x4*)x;

    float a0 = 0.f, a1 = 0.f, a2 = 0.f, a3 = 0.f;
    const int stride = gridDim.x * blockDim.x;
    for (int i = blockIdx.x * blockDim.x + threadIdx.x; i < n4; i += stride) {
        // Streamed once: non-temporal b128 loads keep the 400 MB stream out of
        // the way of anything resident in L2.
        i32x4 sv = __builtin_nontemporal_load(&s4[i]);
        i32x4 ov = __builtin_nontemporal_load(&o4[i]);
        f32x4 xv = __builtin_nontemporal_load(&x4[i]);
        a0 += (sv.x == ov.x) ? xv.x : -xv.x;
        a1 += (sv.y == ov.y) ? xv.y : -xv.y;
        a2 += (sv.z == ov.z) ? xv.z : -xv.z;
        a3 += (sv.w == ov.w) ? xv.w : -xv.w;
    }
    float acc = (a0 + a1) + (a2 + a3);

    // Scalar tail (empty for N % 4 == 0; kept for generality, deterministic).
    if (blockIdx.x == 0 && threadIdx.x == 0) {
        for (int i = n4 * 4; i < n; ++i)
            acc += (s[i] == o[i]) ? x[i] : -x[i];
    }

    // Uniform control flow here: EXEC is all-ones as WMMA requires.
    acc = wave_reduce_sum(acc);

    __shared__ float wsum[NTHREADS / 32];
    const int lane = threadIdx.x & 31;
    const int wid  = threadIdx.x >> 5;
    if (lane == 0) wsum[wid] = acc;
    __syncthreads();
    if (threadIdx.x == 0) {
        float t = 0.f;
        #pragma unroll
        for (int w = 0; w < NTHREADS / 32; ++w) t += wsum[w];
        partials[blockIdx.x] = t;   // fixed order -> deterministic
    }
}

__global__ void __launch_bounds__(NTHREADS)
disc_loss_final_kernel(const float* __restrict__ partials,
                       float* __restrict__ out,
                       int nparts, float inv_n) {
    __shared__ float sm[NTHREADS];
    float a = 0.f;
    for (int i = threadIdx.x; i < nparts; i += NTHREADS)
        a += partials[i];                       // fixed per-thread order
    sm[threadIdx.x] = a;
    __syncthreads();
    #pragma unroll
    for (int s = NTHREADS / 2; s > 0; s >>= 1) {
        if (threadIdx.x < s) sm[threadIdx.x] += sm[threadIdx.x + s];
        __syncthreads();
    }
    if (threadIdx.x == 0) out[0] = sm[0] * inv_n;  // unconditional overwrite
}

extern "C" void kernel_launch(void* const* d_in, const int* in_sizes, int n_in,
                              void* d_out, int out_size, void* d_ws, size_t ws_size,
                              hipStream_t stream) {
    const int*   s = (const int*)d_in[0];
    const int*   o = (const int*)d_in[1];
    const float* x = (const float*)d_in[2];
    const int    n = in_sizes[0];

    float* partials = (float*)d_ws;   // NBLOCKS floats = 8 KB of scratch

    disc_loss_partial_kernel<<<NBLOCKS, NTHREADS, 0, stream>>>(
        s, o, x, partials, n / 4, n);
    disc_loss_final_kernel<<<1, NTHREADS, 0, stream>>>(
        partials, (float*)d_out, NBLOCKS, 1.0f / (float)n);
}